// MultiViewTripletLoss_26594437497379
// MI455X (gfx1250) — compile-verified
//
#include <hip/hip_runtime.h>
#include <hip/hip_bf16.h>

typedef __attribute__((ext_vector_type(2))) float v2f;
typedef __attribute__((ext_vector_type(8))) float v8f;

// Explicit global address space so the hot-loop B loads lower to
// global_load_b64 (LOADcnt only) instead of flat_load_b64 (LOADcnt+DScnt).
#define AS_GLOBAL __attribute__((address_space(1)))
typedef const AS_GLOBAL float* gcfp;
typedef const AS_GLOBAL v2f*   gv2fp;

#define B_ITEMS 4096
#define DDIM    256
#define NCOLS   (3 * B_ITEMS)     // 12288 columns: [A | P0 | P1]
#define NTILES  (NCOLS / 16)      // 768 16-wide column tiles
#define NQUADS  (NTILES / 4)      // 192 quads of 4 tiles
#define CSPLIT  12                // column splits -> grid = 256 x 12 blocks
#define MARGINF 1.0f
#define EPSX    1e-12f
#define SA_STRIDE 260             // 256 + 4 pad: conflict-free b64 LDS reads

__device__ __forceinline__ unsigned encf(float f) {
  unsigned u = __float_as_uint(f);
  return (u & 0x80000000u) ? ~u : (u | 0x80000000u);   // monotone float->uint
}
__device__ __forceinline__ float decf(unsigned u) {
  return __uint_as_float((u & 0x80000000u) ? (u ^ 0x80000000u) : ~u);
}

// ---------------------------------------------------------------------------
// Kernel 1: inverse norms for all 12288 rows; also zero-init encoded maxes.
// One wave32 per row.
// ---------------------------------------------------------------------------
__global__ __launch_bounds__(256) void norms_kernel(
    const float* __restrict__ anchor, const float* __restrict__ positive,
    float* __restrict__ invA, float* __restrict__ invP,
    unsigned* __restrict__ maxEnc)
{
  const int gtid = blockIdx.x * blockDim.x + threadIdx.x;
  if (gtid < 3 * B_ITEMS) maxEnc[gtid] = 0u;   // floor below enc(-inf)

  const int gwave = gtid >> 5;
  const int lane  = threadIdx.x & 31;
  if (gwave >= 3 * B_ITEMS) return;

  gcfp src = (gwave < B_ITEMS)
                 ? (gcfp)(anchor + (size_t)gwave * DDIM)
                 : (gcfp)(positive + (size_t)(gwave - B_ITEMS) * DDIM);
  float s = 0.f;
#pragma unroll
  for (int j = 0; j < DDIM / 32; ++j) { float x = src[lane + 32 * j]; s += x * x; }
#pragma unroll
  for (int m = 16; m >= 1; m >>= 1) s += __shfl_xor(s, m, 32);
  const float inv = 1.0f / fmaxf(sqrtf(s), 1e-12f);
  if (lane == 0) {
    if (gwave < B_ITEMS) invA[gwave] = inv;
    else                 invP[gwave - B_ITEMS] = inv;
  }
}

// ---------------------------------------------------------------------------
// Kernel 2: fused gram + masked row-max + diagonal capture.
// grid.x = 256 row tiles (16 anchor rows), grid.y = CSPLIT column splits.
// Each wave: quads of 4 column tiles, 64 x v_wmma_f32_16x16x4_f32 per tile.
// ---------------------------------------------------------------------------
__global__ __launch_bounds__(256) void gemm_max_kernel(
    const float* __restrict__ anchor, const float* __restrict__ positive,
    const int* __restrict__ labels,
    const float* __restrict__ invA, const float* __restrict__ invP,
    unsigned* __restrict__ maxEnc, float* __restrict__ diag)
{
  __shared__ float sA[16 * SA_STRIDE];
  __shared__ int   sLab[16];
  __shared__ float sInv[16];

  const int tid     = threadIdx.x;
  const int lane    = tid & 31;
  const int wave    = tid >> 5;
  const int rowBase = blockIdx.x * 16;

  // Stage this block's 16 anchor rows (16 KB) into LDS, padded stride.
  {
    gcfp aSrc = (gcfp)(anchor + (size_t)rowBase * DDIM);
    for (int idx = tid; idx < 16 * DDIM; idx += 256) {
      const int r = idx >> 8, c = idx & 255;
      sA[r * SA_STRIDE + c] = aSrc[idx];
    }
  }
  if (tid < 16) { sLab[tid] = labels[rowBase + tid]; sInv[tid] = invA[rowBase + tid]; }
  __syncthreads();

  const int half = lane >> 4;        // 0: rows M=0..7, 1: rows M=8..15
  const int ln16 = lane & 15;        // column within tile / row within A frag
  const int koff = half * 2;         // K sub-offset for f32 WMMA fragments
  const float* aRow = &sA[ln16 * SA_STRIDE];

  // Hoist this half's row labels / inv-norms into registers.
  int   rLab[8]; float rInv[8];
#pragma unroll
  for (int r = 0; r < 8; ++r) { rLab[r] = sLab[half * 8 + r]; rInv[r] = sInv[half * 8 + r]; }

  float m0[8], m1[8], m2[8];         // running masked maxes: A / P0 / P1 regions
#pragma unroll
  for (int r = 0; r < 8; ++r) { m0[r] = -__builtin_inff(); m1[r] = m0[r]; m2[r] = m0[r]; }

  for (int q = blockIdx.y * 8 + wave; q < NQUADS; q += gridDim.y * 8) {
    const int colBase  = q * 64;              // 64-aligned; never straddles regions
    const int region   = colBase >> 12;       // 0=A, 1=P0, 2=P1
    const int itemBase = colBase - (region << 12);

    gcfp cptr[4];
    int colLab[4]; float colInv[4];
#pragma unroll
    for (int t = 0; t < 4; ++t) {
      const int item = itemBase + t * 16 + ln16;
      if (region == 0) {
        cptr[t]   = (gcfp)(anchor + (size_t)item * DDIM) + koff;
        colInv[t] = invA[item];
      } else {
        const int pr = item * 2 + (region - 1);
        cptr[t]   = (gcfp)(positive + (size_t)pr * DDIM) + koff;
        colInv[t] = invP[pr];
      }
      colLab[t] = labels[item];
    }

    v8f zero = {};
    v8f acc4[4];
#pragma unroll
    for (int t = 0; t < 4; ++t) acc4[t] = zero;

    // K loop: one LDS A fragment shared by 4 global B tiles per step.
#pragma unroll 8
    for (int k = 0; k < DDIM; k += 4) {
      const v2f a = *(const v2f*)(aRow + k + koff);
#pragma unroll
      for (int t = 0; t < 4; ++t) {
        const v2f b = *(gv2fp)(cptr[t] + k);
        acc4[t] = __builtin_amdgcn_wmma_f32_16x16x4_f32(
            false, a, false, b, (short)0, acc4[t], false, false);
      }
    }

    // Scale to cosine, mask by label, fold into running max; catch diagonals.
#pragma unroll
    for (int t = 0; t < 4; ++t) {
      const int item = itemBase + t * 16 + ln16;
#pragma unroll
      for (int r = 0; r < 8; ++r) {
        const float d  = acc4[t][r] * colInv[t] * rInv[r];
        const float nd = (colLab[t] != rLab[r]) ? d : -__builtin_inff();
        if (region == 0)      m0[r] = fmaxf(m0[r], nd);
        else if (region == 1) m1[r] = fmaxf(m1[r], nd);
        else                  m2[r] = fmaxf(m2[r], nd);
        if (region > 0 && item == rowBase + half * 8 + r)
          diag[(region - 1) * B_ITEMS + item] = d;   // pos_dist dot
      }
    }
  }

  // Butterfly max across the 16 lanes of each half (per-row reduction).
#pragma unroll
  for (int m = 1; m < 16; m <<= 1) {
#pragma unroll
    for (int r = 0; r < 8; ++r) {
      m0[r] = fmaxf(m0[r], __shfl_xor(m0[r], m, 16));
      m1[r] = fmaxf(m1[r], __shfl_xor(m1[r], m, 16));
      m2[r] = fmaxf(m2[r], __shfl_xor(m2[r], m, 16));
    }
  }
  if (ln16 == 0) {
#pragma unroll
    for (int r = 0; r < 8; ++r) {
      const int row = rowBase + half * 8 + r;
      atomicMax(&maxEnc[row],               encf(m0[r]));
      atomicMax(&maxEnc[B_ITEMS + row],     encf(m1[r]));
      atomicMax(&maxEnc[2 * B_ITEMS + row], encf(m2[r]));
    }
  }
}

// ---------------------------------------------------------------------------
// Kernel 3: per-row losses + mean (single block, deterministic tree reduce).
// ---------------------------------------------------------------------------
__global__ __launch_bounds__(256) void finalize_kernel(
    const unsigned* __restrict__ maxEnc, const float* __restrict__ diag,
    float* __restrict__ out)
{
  __shared__ float red[256];
  float acc = 0.f;
  for (int i = threadIdx.x; i < B_ITEMS; i += 256) {
    const float mA  = decf(maxEnc[i]);
    const float mP0 = decf(maxEnc[B_ITEMS + i]);
    const float mP1 = decf(maxEnc[2 * B_ITEMS + i]);
    const float d0  = diag[i];
    const float d1  = diag[B_ITEMS + i];
    const float pos0 = sqrtf(fmaxf(2.f - 2.f * d0, EPSX));
    const float pos1 = sqrtf(fmaxf(2.f - 2.f * d1, EPSX));
    const float neg0 = sqrtf(fmaxf(2.f - 2.f * fmaxf(mA, mP0), EPSX));
    const float neg1 = sqrtf(fmaxf(2.f - 2.f * fmaxf(mA, mP1), EPSX));
    acc += fmaxf(pos0 - neg0 + MARGINF, 0.f) + fmaxf(pos1 - neg1 + MARGINF, 0.f);
  }
  red[threadIdx.x] = acc;
  __syncthreads();
  for (int s = 128; s > 0; s >>= 1) {
    if (threadIdx.x < s) red[threadIdx.x] += red[threadIdx.x + s];
    __syncthreads();
  }
  if (threadIdx.x == 0) out[0] = red[0] / (2.0f * B_ITEMS);
}

// ---------------------------------------------------------------------------
extern "C" void kernel_launch(void* const* d_in, const int* in_sizes, int n_in,
                              void* d_out, int out_size, void* d_ws, size_t ws_size,
                              hipStream_t stream)
{
  const float* anchor   = (const float*)d_in[0];   // [4096, 256]
  const float* positive = (const float*)d_in[1];   // [4096, 2, 256]
  const int*   labels   = (const int*)d_in[2];     // [4096]
  float* out = (float*)d_out;

  // Workspace layout (128 KB total):
  float*    invA   = (float*)d_ws;                       // 4096
  float*    invP   = invA + B_ITEMS;                     // 8192
  unsigned* maxEnc = (unsigned*)(invP + 2 * B_ITEMS);    // 3*4096
  float*    diag   = (float*)(maxEnc + 3 * B_ITEMS);     // 2*4096

  norms_kernel<<<dim3((3 * B_ITEMS) / 8), dim3(256), 0, stream>>>(
      anchor, positive, invA, invP, maxEnc);

  gemm_max_kernel<<<dim3(B_ITEMS / 16, CSPLIT), dim3(256), 0, stream>>>(
      anchor, positive, labels, invA, invP, maxEnc, diag);

  finalize_kernel<<<dim3(1), dim3(256), 0, stream>>>(maxEnc, diag, out);
}